// GNNPolicy_18734647345319
// MI455X (gfx1250) — compile-verified
//
#include <hip/hip_runtime.h>

typedef float v2f __attribute__((ext_vector_type(2)));
typedef float v8f __attribute__((ext_vector_type(8)));

#define HDIM 64
#define NGRAPH 64

// ---------------------------------------------------------------- utilities
__global__ void zero_f32(float* __restrict__ p, long n) {
  long i = (long)blockIdx.x * blockDim.x + threadIdx.x;
  if (i < n) p[i] = 0.0f;
}

// deg[col[e]] += |ew[e]|   (deg stored in `dis` buffer, zeroed beforehand)
__global__ void deg_kernel(const int* __restrict__ col, const float* __restrict__ ew,
                           float* __restrict__ deg, int E) {
  int e = blockIdx.x * blockDim.x + threadIdx.x;
  if (e < E) atomicAdd(&deg[col[e]], fabsf(ew[e]));
}

// in place: dis[i] = rsqrt(deg[i] + 1)
__global__ void dis_kernel(float* __restrict__ d, int N) {
  int i = blockIdx.x * blockDim.x + threadIdx.x;
  if (i < N) d[i] = rsqrtf(d[i] + 1.0f);
}

// ------------------------------------------------- WMMA GEMM: C[N,64] = A[N,K] @ W[K,64]
// One wave computes a 16x64 output stripe. W staged in LDS, pair-interleaved:
//   sW[((k>>1)*HDIM + n)*2 + (k&1)] = W[k][n]
// so each B fragment (W[k][n], W[k+1][n]) is one aligned ds_load_b64 into an
// even VGPR pair, feeding v_wmma_f32_16x16x4_f32 with no fixup movs.
// A-fragment layout (16x4 f32): lanes 0-15 hold M=lane,K={k0,k0+1}; lanes 16-31 K={k0+2,k0+3}.
__global__ void gemm_wmma_f32(const float* __restrict__ A, const float* __restrict__ W,
                              float* __restrict__ C, int N, int K) {
  __shared__ float sW[128 * HDIM];
  const int tid = threadIdx.x;
  for (int i = tid; i < K * HDIM; i += blockDim.x) {
    const int k = i / HDIM, n = i - k * HDIM;
    sW[(((k >> 1) * HDIM + n) << 1) + (k & 1)] = W[i];
  }
  __syncthreads();

  const int wave = tid >> 5;
  const int lane = tid & 31;
  const int m0 = (blockIdx.x * 8 + wave) * 16;
  if (m0 >= N) return;  // whole-wave uniform: EXEC stays all-ones for WMMA

  const int lh = lane & 15;   // 0..15
  const int hi = lane >> 4;   // 0 or 1

  v8f c0 = {}, c1 = {}, c2 = {}, c3 = {};
  const float* arow = A + (long)(m0 + lh) * K + hi * 2;
  // LDS base for this lane: pair index advances by 2 per k-step of 4
  const float* bbase = &sW[((hi * HDIM) + lh) << 1];

#pragma unroll 4
  for (int k0 = 0; k0 < K; k0 += 4) {
    const float2 a2 = *(const float2*)(arow + k0);
    v2f a; a.x = a2.x; a.y = a2.y;

    const float* bp = bbase + ((k0 >> 1) * HDIM << 1);
    const v2f b0 = *(const v2f*)(bp + 0);    // n-tile 0: +16*2 per tile
    const v2f b1 = *(const v2f*)(bp + 32);
    const v2f b2 = *(const v2f*)(bp + 64);
    const v2f b3 = *(const v2f*)(bp + 96);

    c0 = __builtin_amdgcn_wmma_f32_16x16x4_f32(false, a, false, b0, (short)0, c0, false, false);
    c1 = __builtin_amdgcn_wmma_f32_16x16x4_f32(false, a, false, b1, (short)0, c1, false, false);
    c2 = __builtin_amdgcn_wmma_f32_16x16x4_f32(false, a, false, b2, (short)0, c2, false, false);
    c3 = __builtin_amdgcn_wmma_f32_16x16x4_f32(false, a, false, b3, (short)0, c3, false, false);
  }

  // C/D layout: VGPR r -> row m0 + r + hi*8, col = ntile*16 + lh
  float* crow = C + (long)m0 * HDIM;
#pragma unroll
  for (int r = 0; r < 8; ++r) {
    const long rr = (long)(r + hi * 8) * HDIM;
    crow[rr + 0  + lh] = c0[r];
    crow[rr + 16 + lh] = c1[r];
    crow[rr + 32 + lh] = c2[r];
    crow[rr + 48 + lh] = c3[r];
  }
}

// ------------------------------------------------- edge scatter: agg[col] += norm * xw[row]
// 16 lanes per edge, one float4 per lane -> coalesced 256B row gather, 4 L2 atomics per lane.
__global__ void edge_scatter(const int* __restrict__ row, const int* __restrict__ col,
                             const float* __restrict__ ew, const float* __restrict__ dis,
                             const float* __restrict__ xw, float* __restrict__ agg, int E) {
  long tid = (long)blockIdx.x * blockDim.x + threadIdx.x;
  if (tid >= (long)E * 16) return;
  const int e = (int)(tid >> 4);
  const int c = (int)(tid & 15);
  const int r = row[e], cl = col[e];
  const float nw = dis[r] * fabsf(ew[e]) * dis[cl];
  const float4 v = ((const float4*)(xw + (long)r * HDIM))[c];
  float* out = agg + (long)cl * HDIM + c * 4;
  atomicAdd(out + 0, nw * v.x);
  atomicAdd(out + 1, nw * v.y);
  atomicAdd(out + 2, nw * v.z);
  atomicAdd(out + 3, nw * v.w);
}

// ------------------------------------------------- h = relu(agg + dis^2 * xw + b), in place into agg
__global__ void self_bias_relu(float* __restrict__ agg, const float* __restrict__ xw,
                               const float* __restrict__ dis, const float* __restrict__ b, int N) {
  long tid = (long)blockIdx.x * blockDim.x + threadIdx.x;
  if (tid >= (long)N * 16) return;
  const int i = (int)(tid >> 4);
  const int c = (int)(tid & 15);
  float d2 = dis[i]; d2 *= d2;
  float4 a = ((float4*)agg)[tid];
  const float4 x = ((const float4*)xw)[tid];
  const float4 bb = ((const float4*)b)[c];
  a.x = fmaxf(a.x + d2 * x.x + bb.x, 0.0f);
  a.y = fmaxf(a.y + d2 * x.y + bb.y, 0.0f);
  a.z = fmaxf(a.z + d2 * x.z + bb.z, 0.0f);
  a.w = fmaxf(a.w + d2 * x.w + bb.w, 0.0f);
  ((float4*)agg)[tid] = a;
}

// ------------------------------------------------- final layer fused: h2, actor logits, pooling sums
// One wave32 per node; 2 features per lane.
__global__ void head_kernel(const float* __restrict__ agg2, const float* __restrict__ xw2,
                            const float* __restrict__ dis, const float* __restrict__ b2,
                            const float* __restrict__ actor_w, const float* __restrict__ actor_b,
                            const int* __restrict__ batch, float* __restrict__ logits,
                            float* __restrict__ sums, float* __restrict__ counts, int N) {
  const int i = (int)(((long)blockIdx.x * blockDim.x + threadIdx.x) >> 5);
  const int lane = threadIdx.x & 31;
  if (i >= N) return;
  float d2 = dis[i]; d2 *= d2;
  const int f0 = lane, f1 = lane + 32;
  const long base = (long)i * HDIM;
  const float h0 = fmaxf(agg2[base + f0] + d2 * xw2[base + f0] + b2[f0], 0.0f);
  const float h1 = fmaxf(agg2[base + f1] + d2 * xw2[base + f1] + b2[f1], 0.0f);
  float p = h0 * actor_w[f0] + h1 * actor_w[f1];
#pragma unroll
  for (int off = 16; off > 0; off >>= 1) p += __shfl_xor(p, off, 32);
  const int g = batch[i];
  atomicAdd(&sums[g * HDIM + f0], h0);
  atomicAdd(&sums[g * HDIM + f1], h1);
  if (lane == 0) {
    logits[i] = p + actor_b[0];
    atomicAdd(&counts[g], 1.0f);
  }
}

// ------------------------------------------------- critic: value[g] = mean_embed . critic_w + b
__global__ void value_kernel(const float* __restrict__ sums, const float* __restrict__ counts,
                             const float* __restrict__ critic_w, const float* __restrict__ critic_b,
                             float* __restrict__ value) {
  const int g = threadIdx.x;
  if (g >= NGRAPH) return;
  const float inv = 1.0f / fmaxf(counts[g], 1.0f);
  float acc = 0.0f;
  for (int f = 0; f < HDIM; ++f) acc += sums[g * HDIM + f] * inv * critic_w[f];
  value[g] = acc + critic_b[0];
}

// ----------------------------------------------------------------------------
extern "C" void kernel_launch(void* const* d_in, const int* in_sizes, int n_in,
                              void* d_out, int out_size, void* d_ws, size_t ws_size,
                              hipStream_t stream) {
  const float* x        = (const float*)d_in[0];
  const int*   eidx     = (const int*)  d_in[1];
  const float* ew       = (const float*)d_in[2];
  const int*   batch    = (const int*)  d_in[3];
  const float* W1       = (const float*)d_in[4];
  const float* b1       = (const float*)d_in[5];
  const float* W2       = (const float*)d_in[6];
  const float* b2       = (const float*)d_in[7];
  const float* actor_w  = (const float*)d_in[8];
  const float* actor_b  = (const float*)d_in[9];
  const float* critic_w = (const float*)d_in[10];
  const float* critic_b = (const float*)d_in[11];

  const int Hh = in_sizes[5];          // 64
  const int F  = in_sizes[4] / Hh;     // 128
  const int N  = in_sizes[0] / F;      // 100000
  const int E  = in_sizes[2];          // 1600000
  const int* row = eidx;
  const int* col = eidx + E;

  float* ws     = (float*)d_ws;
  float* dis    = ws;                         // N   (deg, then rsqrt in place)
  float* xw     = dis  + N;                   // N*H (xw1, then reused as xw2)
  float* agg1   = xw   + (long)N * Hh;        // N*H (-> h1 in place)
  float* agg2   = agg1 + (long)N * Hh;        // N*H
  float* sums   = agg2 + (long)N * Hh;        // G*H
  float* counts = sums + NGRAPH * Hh;         // G
  const long total_ws = (long)N + 3L * N * Hh + NGRAPH * Hh + NGRAPH;

  float* logits = (float*)d_out;              // [N]
  float* value  = (float*)d_out + N;          // [G]

  const int T = 256;
  // 1) zero all workspace we accumulate into
  zero_f32<<<(unsigned)((total_ws + T - 1) / T), T, 0, stream>>>(ws, total_ws);
  // 2) degree + norm
  deg_kernel<<<(E + T - 1) / T, T, 0, stream>>>(col, ew, dis, E);
  dis_kernel<<<(N + T - 1) / T, T, 0, stream>>>(dis, N);
  // 3) xw1 = x @ W1   (N/16 waves, 8 waves per block)
  {
    int waves = (N + 15) / 16;
    gemm_wmma_f32<<<(waves + 7) / 8, 256, 0, stream>>>(x, W1, xw, N, F);
  }
  // 4) layer-1 scatter + self-loop/bias/relu
  edge_scatter<<<(unsigned)(((long)E * 16 + T - 1) / T), T, 0, stream>>>(row, col, ew, dis, xw, agg1, E);
  self_bias_relu<<<(unsigned)(((long)N * 16 + T - 1) / T), T, 0, stream>>>(agg1, xw, dis, b1, N);
  // 5) xw2 = h1 @ W2  (reuse xw buffer)
  {
    int waves = (N + 15) / 16;
    gemm_wmma_f32<<<(waves + 7) / 8, 256, 0, stream>>>(agg1, W2, xw, N, Hh);
  }
  // 6) layer-2 scatter
  edge_scatter<<<(unsigned)(((long)E * 16 + T - 1) / T), T, 0, stream>>>(row, col, ew, dis, xw, agg2, E);
  // 7) fused h2 + actor head + pooling
  head_kernel<<<(N + 7) / 8, 256, 0, stream>>>(agg2, xw, dis, b2, actor_w, actor_b,
                                               batch, logits, sums, counts, N);
  // 8) critic head
  value_kernel<<<1, 64, 0, stream>>>(sums, counts, critic_w, critic_b, value);
}